// SimpleGCN2_53128745452228
// MI455X (gfx1250) — compile-verified
//
#include <hip/hip_runtime.h>
#include <math.h>

#define NFEAT_IN 256
#define NHID 16

typedef __attribute__((ext_vector_type(2))) float v2f;
typedef __attribute__((ext_vector_type(8))) float v8f;

// ---------------------------------------------------------------------------
// Degree / normalization
// ---------------------------------------------------------------------------
__global__ void gcn_zero_deg(unsigned* __restrict__ deg, int n) {
  int i = blockIdx.x * blockDim.x + threadIdx.x;
  if (i < n) deg[i] = 0u;
}

__global__ void gcn_count_deg(const int* __restrict__ dst, unsigned* __restrict__ deg, int e) {
  int i = blockIdx.x * blockDim.x + threadIdx.x;
  if (i < e) atomicAdd(&deg[dst[i]], 1u);
}

// in-place: buffer holds uint counts, rewritten as float dinv = rsqrt(deg+1)
__global__ void gcn_dinv(float* __restrict__ dinv, int n) {
  int i = blockIdx.x * blockDim.x + threadIdx.x;
  if (i < n) {
    unsigned d = ((const unsigned*)dinv)[i];
    dinv[i] = rsqrtf((float)d + 1.0f);  // +1 for self-loop
  }
}

// ---------------------------------------------------------------------------
// GEMM1: h[N,16] = x[N,256] @ W1[256,16] via V_WMMA_F32_16X16X4_F32
// One wave -> one 16x16 tile of h. 8 waves / block -> 128 rows per block.
// ---------------------------------------------------------------------------
#define W1_LDS_STRIDE 258  // 256 + 2 pad: distinct LDS banks per output column

__global__ __launch_bounds__(256) void gcn_gemm1(const float* __restrict__ x,
                                                 const float* __restrict__ W1,
                                                 float* __restrict__ h, int n) {
  __shared__ float w1t[16 * W1_LDS_STRIDE];  // transposed: w1t[col][k]
  int tid = threadIdx.x;
  // cooperative transpose-load of W1 (256x16 row-major -> [col][k])
  for (int i = tid; i < NFEAT_IN * NHID; i += 256) {
    int k = i >> 4;   // 0..255
    int c = i & 15;   // 0..15
    w1t[c * W1_LDS_STRIDE + k] = W1[i];
  }
  __syncthreads();

  int wave = tid >> 5;
  int lane = tid & 31;
  int row0 = (blockIdx.x * 8 + wave) * 16;
  if (row0 >= n) return;  // wave-uniform: EXEC stays all-ones for WMMA

  int half = lane >> 4;   // 0: K pair {0,1}, 1: K pair {2,3}
  int m = lane & 15;      // A row (M) for A-frag, output column (N) for B-frag

  const float* arow = x + (size_t)(row0 + m) * NFEAT_IN + half * 2;
  const float* brow = w1t + m * W1_LDS_STRIDE + half * 2;

  v8f c = {};
  for (int k = 0; k < NFEAT_IN; k += 4) {
    v2f a = *(const v2f*)(arow + k);   // global_load_b64, coalesced-ish, 8B aligned
    v2f b = *(const v2f*)(brow + k);   // ds_load_b64, conflict-free via pad
    c = __builtin_amdgcn_wmma_f32_16x16x4_f32(false, a, false, b, (short)0, c, false, false);
  }

  // C/D layout: VGPR r holds row M = r + half*8, column N = m
  float* hout = h + (size_t)(row0 + half * 8) * NHID + m;
#pragma unroll
  for (int r = 0; r < 8; ++r) hout[(size_t)r * NHID] = c[r];
}

// ---------------------------------------------------------------------------
// GEMM2: g[N,16] = relu(a1[N,16] + b1) @ W2[16,16]  (bias+ReLU fused into A-frag)
// ---------------------------------------------------------------------------
__global__ __launch_bounds__(256) void gcn_gemm2(const float* __restrict__ a1,
                                                 const float* __restrict__ b1,
                                                 const float* __restrict__ W2,
                                                 float* __restrict__ g, int n) {
  int tid = threadIdx.x;
  int wave = tid >> 5;
  int lane = tid & 31;
  int row0 = (blockIdx.x * 8 + wave) * 16;
  if (row0 >= n) return;

  int half = lane >> 4;
  int m = lane & 15;

  const float* arow = a1 + (size_t)(row0 + m) * NHID + half * 2;
  v8f c = {};
#pragma unroll
  for (int k = 0; k < NHID; k += 4) {
    int ka = k + half * 2;
    v2f araw = *(const v2f*)(arow + k);
    v2f a;
    a.x = fmaxf(araw.x + b1[ka], 0.0f);
    a.y = fmaxf(araw.y + b1[ka + 1], 0.0f);
    v2f b;
    b.x = W2[ka * 16 + m];        // W2 is 1KB -> L2/L0 resident
    b.y = W2[(ka + 1) * 16 + m];
    c = __builtin_amdgcn_wmma_f32_16x16x4_f32(false, a, false, b, (short)0, c, false, false);
  }

  float* gout = g + (size_t)(row0 + half * 8) * NHID + m;
#pragma unroll
  for (int r = 0; r < 8; ++r) gout[(size_t)r * NHID] = c[r];
}

// ---------------------------------------------------------------------------
// Aggregation: out[i] = h[i]*dinv[i]^2 (self-loop init), then edge scatter-add
// ---------------------------------------------------------------------------
__global__ void gcn_selfloop(const float* __restrict__ hin, const float* __restrict__ dinv,
                             float* __restrict__ out, int n16) {
  int i = blockIdx.x * blockDim.x + threadIdx.x;
  if (i < n16) {
    float di = dinv[i >> 4];
    out[i] = hin[i] * di * di;
  }
}

__global__ void gcn_scatter(const int* __restrict__ src, const int* __restrict__ dst,
                            const float* __restrict__ hin, const float* __restrict__ dinv,
                            float* __restrict__ out, int e) {
  int gid = blockIdx.x * blockDim.x + threadIdx.x;   // e*16 = 51.2M < 2^31
  if (gid >= e * 16) return;
  int c = gid & 15;
  int ed = gid >> 4;
  int s = src[ed];
  int d = dst[ed];
  float v = hin[(size_t)s * 16 + c] * dinv[s] * dinv[d];
  unsafeAtomicAdd(&out[(size_t)d * 16 + c], v);       // global_atomic_add_f32
}

// ---------------------------------------------------------------------------
// Final: out[i,:] = log_softmax(out[i,:] + b2), in place, one thread per node
// ---------------------------------------------------------------------------
__global__ void gcn_logsoftmax(float* __restrict__ out, const float* __restrict__ b2, int n) {
  int i = blockIdx.x * blockDim.x + threadIdx.x;
  if (i >= n) return;
  float t[16];
  float mx = -INFINITY;
#pragma unroll
  for (int c = 0; c < 16; ++c) {
    t[c] = out[(size_t)i * 16 + c] + b2[c];
    mx = fmaxf(mx, t[c]);
  }
  float s = 0.0f;
#pragma unroll
  for (int c = 0; c < 16; ++c) s += expf(t[c] - mx);
  float lse = mx + logf(s);
#pragma unroll
  for (int c = 0; c < 16; ++c) out[(size_t)i * 16 + c] = t[c] - lse;
}

// ---------------------------------------------------------------------------
extern "C" void kernel_launch(void* const* d_in, const int* in_sizes, int n_in,
                              void* d_out, int out_size, void* d_ws, size_t ws_size,
                              hipStream_t stream) {
  const float* x  = (const float*)d_in[0];
  const int*   ei = (const int*)d_in[1];
  const float* W1 = (const float*)d_in[2];
  const float* b1 = (const float*)d_in[3];
  const float* W2 = (const float*)d_in[4];
  const float* b2 = (const float*)d_in[5];

  const int n = in_sizes[0] / NFEAT_IN;  // 100000
  const int e = in_sizes[1] / 2;         // 3200000
  const int* src = ei;
  const int* dst = ei + e;

  // workspace layout (floats): [dinv: n][h: 16n][a1: 16n]; g reuses h
  float* wsf  = (float*)d_ws;
  float* dinv = wsf;
  float* h    = wsf + n;
  float* a1   = h + (size_t)n * 16;
  float* g    = h;                       // h dead after scatter1; reuse for layer-2 products
  float* out  = (float*)d_out;

  const int B = 256;
  const int grid_n    = (n + B - 1) / B;
  const int grid_e    = (e + B - 1) / B;
  const int grid_n16  = (n * 16 + B - 1) / B;
  const int grid_e16  = (int)(((long long)e * 16 + B - 1) / B);
  const int grid_gemm = (n + 127) / 128;  // 8 waves * 16 rows per block

  // normalization: deg -> dinv (in place over same buffer)
  gcn_zero_deg<<<grid_n, B, 0, stream>>>((unsigned*)dinv, n);
  gcn_count_deg<<<grid_e, B, 0, stream>>>(dst, (unsigned*)dinv, e);
  gcn_dinv<<<grid_n, B, 0, stream>>>(dinv, n);

  // layer 1: h = x @ W1 ; a1 = Ahat * h
  gcn_gemm1<<<grid_gemm, B, 0, stream>>>(x, W1, h, n);
  gcn_selfloop<<<grid_n16, B, 0, stream>>>(h, dinv, a1, n * 16);
  gcn_scatter<<<grid_e16, B, 0, stream>>>(src, dst, h, dinv, a1, e);

  // layer 2: g = relu(a1 + b1) @ W2 ; out = Ahat * g ; out = log_softmax(out + b2)
  gcn_gemm2<<<grid_gemm, B, 0, stream>>>(a1, b1, W2, g, n);
  gcn_selfloop<<<grid_n16, B, 0, stream>>>(g, dinv, out, n * 16);
  gcn_scatter<<<grid_e16, B, 0, stream>>>(src, dst, g, dinv, out, e);
  gcn_logsoftmax<<<grid_n, B, 0, stream>>>(out, b2, n);
}